// ValueGGNN_111669150311
// MI455X (gfx1250) — compile-verified
//
#include <hip/hip_runtime.h>
#include <math.h>

// Problem constants (match reference)
#define N_NODES 10000
#define N_EDGES 80000
#define CH      1000
#define OUTC    100
#define BATCHES 16
#define GSTEPS  3

// Padded dims: non-power-of-two strides, exact tile coverage.
#define LDH   1008   // 63 full 16-col tiles (CH padded)
#define LDO   112    // 7 full 16-col tiles (OUTC padded)
#define NROWP 10080  // 315 row-pair jobs * 32 rows (N_NODES padded)
#define KPAIRS 500   // CH/2 K-pairs in pair-interleaved B packs

typedef __attribute__((ext_vector_type(2))) float v2f;
typedef __attribute__((ext_vector_type(8))) float v8f;

// ---------------------------------------------------------------------------
// WMMA f32 16x16x4: D = A(16x4)*B(4x16)+C. fp32 kept deliberately: the net is
// scatter/gather bandwidth-bound (~2.5 GB vs 0.42 TFLOP), so full-precision
// matrix math costs nothing and keeps the C=1000 reductions feeding
// sigmoid/tanh numerically tight.
// ---------------------------------------------------------------------------
__device__ __forceinline__ v8f wmma4(v2f a, v2f b, v8f c) {
  return __builtin_amdgcn_wmma_f32_16x16x4_f32(
      false, a, false, b, (short)0, c, false, false);
}

// ---------------------------------------------------------------------------
// Kernel 1: m = h @ Wp.   Wp is pair-interleaved: (pair p, col j) -> float2
// {W[2p][j], W[2p+1][j]} at Wp[(p*LDH+j)*2], so a B-fragment is ONE b64 load.
// 160 thr = 5 waves; wave job = bx*5+w owns 32 rows (2 row-tiles), 112 cols:
// per K-step: 2 A b64 + 7 B b64 -> 14 WMMA. Guard-free.
// ---------------------------------------------------------------------------
__global__ void __launch_bounds__(160)
gemm_m_kernel(const float* __restrict__ h, const float* __restrict__ Wp,
              float* __restrict__ m) {
  const int lane = threadIdx.x & 31;
  const int wave = threadIdx.x >> 5;
  const int job  = blockIdx.x * 5 + wave;     // 0..314
  const int M0   = job * 32;
  const int N0   = blockIdx.y * 112;
  const int mr   = lane & 15;
  const int pj   = lane >> 4;                  // K-pair select (0/1)

  const float* ap = h  + (long)(M0 + mr) * LDH + 2 * pj;
  const float* bp = Wp + ((long)pj * LDH + N0 + mr) * 2;

  v8f acc[2][7] = {};
  for (int k = 0; k < CH; k += 4) {
    v2f a0 = *(const v2f*)ap;
    v2f a1 = *(const v2f*)(ap + (long)16 * LDH);
#pragma unroll
    for (int t = 0; t < 7; ++t) {
      v2f b = *(const v2f*)(bp + 32 * t);
      acc[0][t] = wmma4(a0, b, acc[0][t]);
      acc[1][t] = wmma4(a1, b, acc[1][t]);
    }
    ap += 4;
    bp += 4 * LDH;        // 2 pairs forward
  }

  const int half = pj << 3;
  float* dp = m + (long)M0 * LDH + N0 + mr;
#pragma unroll
  for (int tt = 0; tt < 2; ++tt)
#pragma unroll
    for (int t = 0; t < 7; ++t)
#pragma unroll
      for (int r = 0; r < 8; ++r)
        dp[(long)(16 * tt + r + half) * LDH + 16 * t] = acc[tt][t][r];
}

// ---------------------------------------------------------------------------
// Kernel 2: edge scatter: agg[dst[e]] += m[src[e]] * ew[e] (f32 atomics, L2)
// ---------------------------------------------------------------------------
__global__ void scatter_kernel(const float* __restrict__ m, const float* __restrict__ ew,
                               const int* __restrict__ src, const int* __restrict__ dst,
                               float* __restrict__ agg) {
  const int e = blockIdx.x;
  const int s = src[e];
  const int d = dst[e];
  const float w = ew[e];
  const float* ms = m + (long)s * LDH;
  float* ad = agg + (long)d * LDH;
  for (int c = threadIdx.x; c < CH; c += blockDim.x)
    atomicAdd(&ad[c], ms[c] * w);
}

// ---------------------------------------------------------------------------
// Kernel 3: fully fused GRU. Wave -> 32 rows x one 16-col tile of ALL SIX
// gate pre-activations (12 accumulators), gates applied in-register, h_next
// written directly. Bih/Bhh are transposed + pair-interleaved, gates at
// +g*CH*LDH floats (fits 24-bit imm offsets -> single address reg each).
// Per K-step: 4 A b64 + 6 B b64 -> 12 WMMA. Guard-free K-loop.
// ---------------------------------------------------------------------------
__global__ void __launch_bounds__(160)
gru_fused_kernel(const float* __restrict__ agg, const float* __restrict__ h,
                 const float* __restrict__ Bih, const float* __restrict__ Bhh,
                 const float* __restrict__ b_ih, const float* __restrict__ b_hh,
                 float* __restrict__ h_next) {
  const int lane = threadIdx.x & 31;
  const int wave = threadIdx.x >> 5;
  const int job  = blockIdx.x * 5 + wave;
  const int M0   = job * 32;
  const int J0   = blockIdx.y * 16;
  const int mr   = lane & 15;
  const int pj   = lane >> 4;
  const long GSZ = (long)CH * LDH;             // one gate matrix (floats)

  const float* aA = agg + (long)(M0 + mr) * LDH + 2 * pj;
  const float* aH = h   + (long)(M0 + mr) * LDH + 2 * pj;
  const float* pih = Bih + ((long)pj * LDH + J0 + mr) * 2;
  const float* phh = Bhh + ((long)pj * LDH + J0 + mr) * 2;

  v8f ai[2][3] = {};   // [row-tile][gate r,z,n]  from agg @ w_ih^T
  v8f ah[2][3] = {};   //                          from h   @ w_hh^T

  for (int k = 0; k < CH; k += 4) {
    v2f A0 = *(const v2f*)aA;
    v2f A1 = *(const v2f*)(aA + (long)16 * LDH);
    v2f H0 = *(const v2f*)aH;
    v2f H1 = *(const v2f*)(aH + (long)16 * LDH);
#pragma unroll
    for (int g = 0; g < 3; ++g) {
      v2f Bi = *(const v2f*)(pih + g * GSZ);
      v2f Bh = *(const v2f*)(phh + g * GSZ);
      ai[0][g] = wmma4(A0, Bi, ai[0][g]);
      ai[1][g] = wmma4(A1, Bi, ai[1][g]);
      ah[0][g] = wmma4(H0, Bh, ah[0][g]);
      ah[1][g] = wmma4(H1, Bh, ah[1][g]);
    }
    aA += 4; aH += 4;
    pih += 4 * LDH; phh += 4 * LDH;
  }

  const int col = J0 + mr;
  const int half = pj << 3;
  if (col < CH) {                               // single guard, outside K-loop
    const float bir = b_ih[col],          bhr = b_hh[col];
    const float biz = b_ih[CH + col],     bhz = b_hh[CH + col];
    const float bin = b_ih[2 * CH + col], bhn = b_hh[2 * CH + col];
#pragma unroll
    for (int tt = 0; tt < 2; ++tt)
#pragma unroll
      for (int r = 0; r < 8; ++r) {
        const long row = M0 + 16 * tt + r + half;
        const float ir = ai[tt][0][r] + bir, hr = ah[tt][0][r] + bhr;
        const float iz = ai[tt][1][r] + biz, hz = ah[tt][1][r] + bhz;
        const float in_ = ai[tt][2][r] + bin, hn = ah[tt][2][r] + bhn;
        const float rg = 1.0f / (1.0f + __expf(-(ir + hr)));
        const float zg = 1.0f / (1.0f + __expf(-(iz + hz)));
        const float ng = tanhf(in_ + rg * hn);
        const float hp = h[row * LDH + col];
        h_next[row * LDH + col] = (1.0f - zg) * ng + zg * hp;
      }
  }
}

// ---------------------------------------------------------------------------
// Kernel 4: out_h = relu(h) @ fc_w^T + fc_b  (Bfc transposed+pair-interleaved)
// Wave: 32 rows x all 7 output col-tiles.
// ---------------------------------------------------------------------------
__global__ void __launch_bounds__(160)
fc_kernel(const float* __restrict__ h, const float* __restrict__ Bfc,
          const float* __restrict__ fc_b, float* __restrict__ out_h) {
  const int lane = threadIdx.x & 31;
  const int wave = threadIdx.x >> 5;
  const int job  = blockIdx.x * 5 + wave;
  const int M0   = job * 32;
  const int mr   = lane & 15;
  const int pj   = lane >> 4;

  const float* ap = h + (long)(M0 + mr) * LDH + 2 * pj;
  const float* bp = Bfc + ((long)pj * LDO + mr) * 2;

  v8f acc[2][7] = {};
  for (int k = 0; k < CH; k += 4) {
    v2f a0 = *(const v2f*)ap;
    v2f a1 = *(const v2f*)(ap + (long)16 * LDH);
    a0.x = fmaxf(a0.x, 0.0f); a0.y = fmaxf(a0.y, 0.0f);   // fused ReLU
    a1.x = fmaxf(a1.x, 0.0f); a1.y = fmaxf(a1.y, 0.0f);
#pragma unroll
    for (int t = 0; t < 7; ++t) {
      v2f b = *(const v2f*)(bp + 32 * t);
      acc[0][t] = wmma4(a0, b, acc[0][t]);
      acc[1][t] = wmma4(a1, b, acc[1][t]);
    }
    ap += 4;
    bp += 4 * LDO;
  }

  const int half = pj << 3;
#pragma unroll
  for (int t = 0; t < 7; ++t) {
    const int col = 16 * t + mr;
    if (col < OUTC) {
      const float bias = fc_b[col];
#pragma unroll
      for (int tt = 0; tt < 2; ++tt)
#pragma unroll
        for (int r = 0; r < 8; ++r)
          out_h[(long)(M0 + 16 * tt + r + half) * LDO + col] = acc[tt][t][r] + bias;
    }
  }
}

// ---------------------------------------------------------------------------
// Per-launch weight packing (~37 MB once vs ~2.5 GB total traffic).
// Pair-interleaved B layout: element (k, j) -> pack[((k/2)*ld + j)*2 + (k&1)].
// ---------------------------------------------------------------------------
__global__ void pack_w_kernel(const float* __restrict__ W, float* __restrict__ Wp,
                              long n) {   // [3][CH][CH] -> [3][KPAIRS][LDH][2]
  long i = (long)blockIdx.x * blockDim.x + threadIdx.x;
  const long stride = (long)gridDim.x * blockDim.x;
  for (; i < n; i += stride) {
    const int f = (int)(i & 1);
    const long r2 = i >> 1;
    const int j = (int)(r2 % LDH);
    const long sp = r2 / LDH;               // s*KPAIRS + p
    const int s = (int)(sp / KPAIRS);
    const int p = (int)(sp % KPAIRS);
    const int k = 2 * p + f;
    Wp[i] = (j < CH) ? W[((long)s * CH + k) * CH + j] : 0.0f;
  }
}

__global__ void transpose_pad_kernel(const float* __restrict__ w, float* __restrict__ Bt,
                                     long n) {  // [3C][C] -> [3][KPAIRS][LDH][2], B=w^T
  long i = (long)blockIdx.x * blockDim.x + threadIdx.x;
  const long stride = (long)gridDim.x * blockDim.x;
  for (; i < n; i += stride) {
    const int f = (int)(i & 1);
    const long r2 = i >> 1;
    const int j = (int)(r2 % LDH);
    const long gp = r2 / LDH;
    const int g = (int)(gp / KPAIRS);
    const int p = (int)(gp % KPAIRS);
    const int k = 2 * p + f;
    Bt[i] = (j < CH) ? w[((long)g * CH + j) * CH + k] : 0.0f;
  }
}

__global__ void transpose_fc_kernel(const float* __restrict__ fc_w,
                                    float* __restrict__ Bfc, long n) { // -> [KPAIRS][LDO][2]
  long i = (long)blockIdx.x * blockDim.x + threadIdx.x;
  const long stride = (long)gridDim.x * blockDim.x;
  for (; i < n; i += stride) {
    const int f = (int)(i & 1);
    const long r2 = i >> 1;
    const int j = (int)(r2 % LDO);
    const int p = (int)(r2 / LDO);
    const int k = 2 * p + f;
    Bfc[i] = (j < OUTC) ? fc_w[(long)j * CH + k] : 0.0f;
  }
}

__global__ void copy_pad_kernel(const float* __restrict__ x, float* __restrict__ h0,
                                long n) {  // [N][C] -> [NROWP][LDH]
  long i = (long)blockIdx.x * blockDim.x + threadIdx.x;
  const long stride = (long)gridDim.x * blockDim.x;
  for (; i < n; i += stride) {
    const long nrow = i / CH;
    const int c = (int)(i % CH);
    h0[nrow * LDH + c] = x[i];
  }
}

__global__ void zero_kernel(float* __restrict__ p, long n) {
  long i = (long)blockIdx.x * blockDim.x + threadIdx.x;
  const long stride = (long)gridDim.x * blockDim.x;
  for (; i < n; i += stride) p[i] = 0.0f;
}

// ---------------------------------------------------------------------------
// Pooling
// ---------------------------------------------------------------------------
__global__ void pool_kernel(const float* __restrict__ out_h, const int* __restrict__ batch,
                            float* __restrict__ sums, float* __restrict__ cnts) {
  const int n = blockIdx.x;
  const int b = batch[n];
  for (int j = threadIdx.x; j < OUTC; j += blockDim.x)
    atomicAdd(&sums[b * OUTC + j], out_h[(long)n * LDO + j]);
  if (threadIdx.x == 0) atomicAdd(&cnts[b], 1.0f);
}

__global__ void finalize_kernel(const float* __restrict__ sums,
                                const float* __restrict__ cnts,
                                float* __restrict__ out) {
  const int b = threadIdx.x;
  if (b < BATCHES) {
    const float c = fmaxf(cnts[b], 1.0f);
    float s = 0.0f;
    for (int j = 0; j < OUTC; ++j) s += sums[b * OUTC + j];
    out[b] = s / (c * (float)OUTC);
  }
}

// ---------------------------------------------------------------------------
// Launcher
// ---------------------------------------------------------------------------
extern "C" void kernel_launch(void* const* d_in, const int* in_sizes, int n_in,
                              void* d_out, int out_size, void* d_ws, size_t ws_size,
                              hipStream_t stream) {
  const float* x     = (const float*)d_in[0];
  const float* ew    = (const float*)d_in[1];
  const float* Wall  = (const float*)d_in[2];   // [3, C, C]
  const float* w_ih  = (const float*)d_in[3];   // [3C, C]
  const float* w_hh  = (const float*)d_in[4];   // [3C, C]
  const float* b_ih  = (const float*)d_in[5];   // [3C]
  const float* b_hh  = (const float*)d_in[6];   // [3C]
  const float* fc_w  = (const float*)d_in[7];   // [OUT, C]
  const float* fc_b  = (const float*)d_in[8];   // [OUT]
  const int*   esrc  = (const int*)d_in[9];
  const int*   edst  = (const int*)d_in[10];
  const int*   batch = (const int*)d_in[11];
  float*       out   = (float*)d_out;

  // Workspace layout (fp32, all padded; offsets are even -> b64 alignment ok)
  float* ws = (float*)d_ws;
  const size_t NLD = (size_t)NROWP * LDH;             // padded activation
  const size_t WSZ = (size_t)3 * CH * LDH;            // packed weight block
  float* h0    = ws;
  float* h1    = h0 + NLD;
  float* mbuf  = h1 + NLD;
  float* agg   = mbuf + NLD;
  float* Wp    = agg + NLD;
  float* Bih   = Wp + WSZ;
  float* Bhh   = Bih + WSZ;
  float* Bfc   = Bhh + WSZ;
  float* out_h = Bfc + (size_t)CH * LDO;
  float* sums  = out_h + (size_t)NROWP * LDO;
  float* cnts  = sums + (size_t)BATCHES * OUTC;

  // ---- per-launch packing ----
  pack_w_kernel<<<1024, 256, 0, stream>>>(Wall, Wp, (long)WSZ);
  transpose_pad_kernel<<<1024, 256, 0, stream>>>(w_ih, Bih, (long)WSZ);
  transpose_pad_kernel<<<1024, 256, 0, stream>>>(w_hh, Bhh, (long)WSZ);
  transpose_fc_kernel<<<256, 256, 0, stream>>>(fc_w, Bfc, (long)CH * LDO);
  copy_pad_kernel<<<2048, 256, 0, stream>>>(x, h0, (long)N_NODES * CH);
  // zero the 80 padded rows of both h buffers (keeps padded lanes finite)
  zero_kernel<<<64, 256, 0, stream>>>(h0 + (size_t)N_NODES * LDH,
                                      (long)(NROWP - N_NODES) * LDH);
  zero_kernel<<<64, 256, 0, stream>>>(h1 + (size_t)N_NODES * LDH,
                                      (long)(NROWP - N_NODES) * LDH);

  const dim3 blk(160);                  // 5 waves; 63*5 = 315 row-pair jobs exact
  const dim3 grid_m(63, 9);             // 9*112 = 1008 cols exact
  const dim3 grid_gru(63, 63);          // 63*16 = 1008 cols exact
  const dim3 grid_fc(63, 1);

  float* hc = h0;
  float* hn = h1;
  for (int s = 0; s < GSTEPS; ++s) {
    gemm_m_kernel<<<grid_m, blk, 0, stream>>>(hc, Wp + (size_t)s * CH * LDH, mbuf);
    zero_kernel<<<1024, 256, 0, stream>>>(agg, (long)NLD);
    scatter_kernel<<<N_EDGES, 256, 0, stream>>>(mbuf, ew, esrc, edst, agg);
    gru_fused_kernel<<<grid_gru, blk, 0, stream>>>(agg, hc, Bih, Bhh, b_ih, b_hh, hn);
    float* t = hc; hc = hn; hn = t;
  }

  fc_kernel<<<grid_fc, blk, 0, stream>>>(hc, Bfc, fc_b, out_h);
  zero_kernel<<<4, 256, 0, stream>>>(sums, (long)(BATCHES * OUTC + BATCHES));
  pool_kernel<<<N_NODES, 128, 0, stream>>>(out_h, batch, sums, cnts);
  finalize_kernel<<<1, 32, 0, stream>>>(sums, cnts, out);
}